// CRF_13443247637158
// MI455X (gfx1250) — compile-verified
//
#include <hip/hip_runtime.h>
#include <hip/hip_bf16.h>
#include <stdint.h>

// ---------------- problem constants (match reference) ----------------
#define K         35          // tagset size
#define KP        36          // padded forward-var (one -1e30 pad slot)
#define T_LEN     16384
#define CH        128         // steps per chunk (tile held in LDS)
#define NCH       (T_LEN/CH)  // 128 chunks
#define START_IDX 33
#define STOP_IDX  34
#define NEG_INF   (-10000.0f)
#define PAD_NEG   (-1.0e30f)

// ---------------- CDNA5 async global->LDS path (guarded) ----------------
#if defined(__HIP_DEVICE_COMPILE__) && \
    __has_builtin(__builtin_amdgcn_global_load_async_to_lds_b128) && \
    __has_builtin(__builtin_amdgcn_s_wait_asynccnt)
#define USE_ASYNC 1
#else
#define USE_ASYNC 0
#endif

typedef int v4i __attribute__((ext_vector_type(4)));

__device__ __forceinline__ void cp16_async(const void* g, void* l) {
#if USE_ASYNC
  // async DMA: 16B per lane, wave32 -> 512B per instruction, tracked by ASYNCcnt
  __builtin_amdgcn_global_load_async_to_lds_b128(
      (__attribute__((address_space(1))) v4i*)(g),
      (__attribute__((address_space(3))) v4i*)(l),
      0, 0);
#else
  *(float4*)l = *(const float4*)g;   // synchronous fallback
#endif
}

#if USE_ASYNC
#define ASYNC_WAIT(n) __builtin_amdgcn_s_wait_asynccnt(n)
#else
#define ASYNC_WAIT(n) ((void)0)
#endif

// =====================================================================
// Kernel A: sequential Viterbi forward pass (1 block, 64 threads = 2 waves)
//   - forward_var lives in LDS (36 floats, slot 35 padded with -1e30)
//   - feats double-buffered in LDS via async global->LDS copies
//   - backpointers (u8) written to global scratch AND kept per-chunk in LDS
//   - per-chunk tag->tag composition maps enable parallel backtrack later
// =====================================================================
__global__ __launch_bounds__(64) void crf_viterbi_forward(
    const float* __restrict__ feats,      // [T_LEN*K]
    const float* __restrict__ trans,      // [K*K], trans[next*K+prev]
    float* __restrict__ out,              // out[0] = path_score
    unsigned char* __restrict__ bpg,      // [T_LEN*K] backpointers
    int* __restrict__ boundaries)         // [NCH] path tag at end of each chunk
{
  __shared__ __align__(16) float fv[KP];                 // forward var (+pad)
  __shared__ __align__(16) float featbuf[2][CH * K];     // double buffer
  __shared__ __align__(16) unsigned char bpc[CH * K];    // current chunk bptrs
  __shared__ unsigned char maps[NCH][K];                 // chunk end-tag -> entry-tag
  __shared__ float term[K];

  const int tid = threadIdx.x;

  // Per-thread transitions row (next = tid) held in VGPRs.
  float trow[K];
  float tstop = 0.0f;
  if (tid < K) {
#pragma unroll
    for (int p = 0; p < K; ++p) trow[p] = trans[tid * K + p];
    tstop = trans[STOP_IDX * K + tid];
  }

  if (tid < K) fv[tid] = (tid == START_IDX) ? 0.0f : NEG_INF;
  if (tid == K) fv[K] = PAD_NEG;   // pad slot never wins the argmax

  // Prefetch chunk 0 (CH*K floats = 1120 float4s; 18/17 async instrs per wave).
  {
    const float* g = feats;
    float* l = featbuf[0];
    for (int i = tid; i < (CH * K) / 4; i += 64)
      cp16_async(g + 4 * i, l + 4 * i);
  }

  for (int c = 0; c < NCH; ++c) {
    if (c + 1 < NCH) {
      // Issue next chunk into the other buffer, then wait until only those
      // remain in flight (completion is in-order => current chunk is ready).
      const float* g = feats + (size_t)(c + 1) * CH * K;
      float* l = featbuf[(c + 1) & 1];
      for (int i = tid; i < (CH * K) / 4; i += 64)
        cp16_async(g + 4 * i, l + 4 * i);
      ASYNC_WAIT(17);
    } else {
      ASYNC_WAIT(0);
    }
    __syncthreads();

    const float* fb = featbuf[c & 1];

    for (int i = 0; i < CH; ++i) {
      float best = PAD_NEG;
      int   barg = 0;
      float nv   = 0.0f;
      if (tid < K) {
        // snapshot forward_var via 9x ds_load_b128
        float fvr[KP];
        const float4* fv4 = (const float4*)fv;
#pragma unroll
        for (int q = 0; q < KP / 4; ++q) {
          float4 v = fv4[q];
          fvr[4*q+0] = v.x; fvr[4*q+1] = v.y; fvr[4*q+2] = v.z; fvr[4*q+3] = v.w;
        }
        // max-plus row: strict '>' keeps FIRST max index (matches jnp.argmax)
#pragma unroll
        for (int p = 0; p < K; ++p) {
          float s = fvr[p] + trow[p];
          if (s > best) { best = s; barg = p; }
        }
        nv = best + fb[i * K + tid];
      }
      __syncthreads();              // all reads of fv done
      if (tid < K) {
        fv[tid] = nv;
        unsigned char b = (unsigned char)barg;
        bpc[i * K + tid] = b;
        bpg[(size_t)(c * CH + i) * K + tid] = b;
      }
      __syncthreads();              // fv update visible for next step
    }

    // Compose this chunk's backtrack map: end tag e -> tag before the chunk.
    if (tid < K) {
      int tag = tid;
      for (int i = CH - 1; i >= 0; --i) tag = bpc[i * K + tag];
      maps[c][tid] = (unsigned char)tag;
    }
    // next iteration's top __syncthreads() protects bpc/featbuf reuse
  }
  __syncthreads();

  // Terminal scores + argmax, then derive chunk-boundary tags from the maps.
  if (tid < K) term[tid] = fv[tid] + tstop;
  __syncthreads();
  if (tid == 0) {
    float bsc = term[0]; int bl = 0;
    for (int n = 1; n < K; ++n)
      if (term[n] > bsc) { bsc = term[n]; bl = n; }
    out[0] = bsc;                      // path_score

    int tag = bl;                      // tag at end of last chunk
    boundaries[NCH - 1] = tag;
    for (int c = NCH - 1; c >= 1; --c) {
      tag = maps[c][tag];              // tag at end of chunk c-1
      boundaries[c - 1] = tag;
    }
  }
}

// =====================================================================
// Kernel B: parallel per-chunk backtrack (NCH blocks).
// Each block async-loads its chunk's backpointers into LDS and chases
// CH dependent LDS reads locally, then writes the path as floats.
// =====================================================================
__global__ __launch_bounds__(64) void crf_viterbi_backtrack(
    const unsigned char* __restrict__ bpg,
    const int* __restrict__ boundaries,
    float* __restrict__ out)            // out[1 + t] = best_path[t]
{
  __shared__ __align__(16) unsigned char bpl[CH * K];
  __shared__ int pathl[CH];

  const int c   = blockIdx.x;
  const int tid = threadIdx.x;

  const unsigned char* g = bpg + (size_t)c * CH * K;   // 4480B, 16B-aligned
  for (int i = tid; i < (CH * K) / 16; i += 64)
    cp16_async(g + 16 * i, bpl + 16 * i);
  ASYNC_WAIT(0);
  __syncthreads();

  if (tid == 0) {
    int tag = boundaries[c];           // path tag at last step of chunk
    pathl[CH - 1] = tag;
    for (int i = CH - 1; i >= 1; --i) {
      tag = bpl[i * K + tag];          // path[t-1] = bp[t][path[t]]
      pathl[i - 1] = tag;
    }
  }
  __syncthreads();

  float* po = out + 1 + (size_t)c * CH;
  for (int i = tid; i < CH; i += 64) po[i] = (float)pathl[i];
}

// =====================================================================
extern "C" void kernel_launch(void* const* d_in, const int* in_sizes, int n_in,
                              void* d_out, int out_size, void* d_ws, size_t ws_size,
                              hipStream_t stream) {
  (void)in_sizes; (void)n_in; (void)out_size; (void)ws_size;

  const float* feats = (const float*)d_in[0];   // [16384, 35] f32
  const float* trans = (const float*)d_in[1];   // [35, 35] f32
  float* out = (float*)d_out;                   // [1 + 16384] f32

  unsigned char* bpg = (unsigned char*)d_ws;                       // 573440 B
  int* boundaries = (int*)((char*)d_ws + (size_t)T_LEN * K);       // 128 ints

  crf_viterbi_forward<<<dim3(1), dim3(64), 0, stream>>>(
      feats, trans, out, bpg, boundaries);
  crf_viterbi_backtrack<<<dim3(NCH), dim3(64), 0, stream>>>(
      bpg, boundaries, out);
}